// HashtableEmbedding_75514114998642
// MI455X (gfx1250) — compile-verified
//
#include <hip/hip_runtime.h>
#include <stdint.h>

// ---------------------------------------------------------------------------
// HashtableEmbedding on MI455X (gfx1250).
// Memory-bound: ~0.9 GB total traffic dominated by the 436 MB row gather.
// Rank assignment = scatter-min + two-level prefix sum (few MB, negligible).
// Gather uses the CDNA5 async global<->LDS data movers (ASYNCcnt path),
// with NT store hints so the streamed output does not evict the ~256 MB
// embedding-row read set from the 192 MB L2.
// ---------------------------------------------------------------------------

#define RAW_ID_SPACE 500000
#define EMB_DIM      128

// --- Phase 1: first_pos[r] = INT_MAX ---------------------------------------
__global__ void he_init_firstpos(int* __restrict__ first_pos, int raw) {
    int i = blockIdx.x * blockDim.x + threadIdx.x;
    if (i < raw) first_pos[i] = 0x7FFFFFFF;
}

// --- Phase 2: scatter-min of position into raw-id space --------------------
__global__ void he_scatter_min(const int* __restrict__ ids,
                               int* __restrict__ first_pos, int n) {
    int i = (blockIdx.x * blockDim.x + threadIdx.x) * 4;
    if (i + 3 < n) {
        int4 v = *(const int4*)(ids + i);
        atomicMin(&first_pos[v.x], i);
        atomicMin(&first_pos[v.y], i + 1);
        atomicMin(&first_pos[v.z], i + 2);
        atomicMin(&first_pos[v.w], i + 3);
    } else {
        for (int k = 0; k < 4 && i + k < n; ++k)
            atomicMin(&first_pos[ids[i + k]], i + k);
    }
}

// --- Phase 3: per-block (1024 positions) count of first occurrences --------
__global__ void he_count_firsts(const int* __restrict__ ids,
                                const int* __restrict__ first_pos,
                                int* __restrict__ block_sums, int n) {
    int t = threadIdx.x;
    int base = blockIdx.x * 1024 + t * 4;
    int cnt = 0;
    if (base + 3 < n) {
        int4 v = *(const int4*)(ids + base);
        cnt += (first_pos[v.x] == base);
        cnt += (first_pos[v.y] == base + 1);
        cnt += (first_pos[v.z] == base + 2);
        cnt += (first_pos[v.w] == base + 3);
    } else {
        for (int k = 0; k < 4; ++k) {
            int p = base + k;
            if (p < n) cnt += (first_pos[ids[p]] == p);
        }
    }
    __shared__ int sh[256];
    sh[t] = cnt;
    __syncthreads();
    for (int off = 128; off > 0; off >>= 1) {
        if (t < off) sh[t] += sh[t + off];
        __syncthreads();
    }
    if (t == 0) block_sums[blockIdx.x] = sh[0];
}

// --- Phase 4: single-block scan of block sums -> exclusive block offsets ---
__global__ void he_scan_blocksums(const int* __restrict__ block_sums,
                                  int* __restrict__ block_off, int nb) {
    __shared__ int sh[1024];
    __shared__ int carry;
    int t = threadIdx.x;
    if (t == 0) carry = 0;
    __syncthreads();
    for (int base = 0; base < nb; base += 1024) {
        int v = (base + t < nb) ? block_sums[base + t] : 0;
        sh[t] = v;
        __syncthreads();
        for (int off = 1; off < 1024; off <<= 1) {   // Hillis-Steele inclusive
            int x = (t >= off) ? sh[t - off] : 0;
            __syncthreads();
            sh[t] += x;
            __syncthreads();
        }
        if (base + t < nb) block_off[base + t] = carry + sh[t] - v;  // exclusive
        __syncthreads();
        if (t == 0) carry += sh[1023];
        __syncthreads();
    }
}

// --- Phase 5: write slot[raw_id] = first-appearance rank -------------------
__global__ void he_write_slots(const int* __restrict__ ids,
                               const int* __restrict__ first_pos,
                               const int* __restrict__ block_off,
                               int* __restrict__ slot, int n) {
    int t = threadIdx.x;
    int base = blockIdx.x * 1024 + t * 4;
    int idv[4], isf[4], cnt = 0;
    for (int k = 0; k < 4; ++k) {
        int p = base + k;
        if (p < n) {
            idv[k] = ids[p];
            isf[k] = (first_pos[idv[k]] == p);
        } else {
            idv[k] = 0;
            isf[k] = 0;
        }
        cnt += isf[k];
    }
    __shared__ int sh[256];
    sh[t] = cnt;
    __syncthreads();
    for (int off = 1; off < 256; off <<= 1) {        // inclusive scan
        int x = (t >= off) ? sh[t - off] : 0;
        __syncthreads();
        sh[t] += x;
        __syncthreads();
    }
    int excl = block_off[blockIdx.x] + sh[t] - cnt;  // exclusive prefix
    for (int k = 0; k < 4; ++k) {
        if (isf[k]) { slot[idv[k]] = excl; ++excl; }
    }
}

// --- Phase 6: gather rows via CDNA5 async global<->LDS movers --------------
// One wave per 8 rows: 32 lanes x 16 B = 512 B = one embedding row.
// Issue 8 async b128 loads (per-lane gather addresses) into this wave's LDS
// region, one s_wait_asynccnt drain, then 8 async NT b128 stores LDS -> out.
// NT on stores keeps the L2 for the reused embedding rows.
__global__ void he_gather_rows(const int* __restrict__ ids,
                               const int* __restrict__ slot,
                               const float* __restrict__ emb,
                               float* __restrict__ out, int n) {
    __shared__ __align__(16) char stage[8 * 8 * 512];   // 8 waves * 8 rows * 512B
    int wave = threadIdx.x >> 5;
    int lane = threadIdx.x & 31;
    int row0 = (blockIdx.x * 8 + wave) * 8;

    if (row0 + 7 < n) {
        unsigned base_off =
            (unsigned)(uintptr_t)stage + (unsigned)(wave * 4096 + lane * 16);
        const float* dsts[8];
#pragma unroll
        for (int k = 0; k < 8; ++k) {
            int row = row0 + k;
            int s = slot[ids[row]];
            const float* src = emb + (size_t)s * EMB_DIM + lane * 4;
            unsigned lo = base_off + (unsigned)(k * 512);
            asm volatile("global_load_async_to_lds_b128 %0, %1, off"
                         :: "v"(lo), "v"(src) : "memory");
            dsts[k] = out + (size_t)row * EMB_DIM + lane * 4;
        }
        asm volatile("s_wait_asynccnt 0" ::: "memory");
#pragma unroll
        for (int k = 0; k < 8; ++k) {
            unsigned lo = base_off + (unsigned)(k * 512);
            asm volatile("global_store_async_from_lds_b128 %0, %1, off th:TH_STORE_NT"
                         :: "v"(dsts[k]), "v"(lo) : "memory");
        }
        asm volatile("s_wait_asynccnt 0" ::: "memory");
    } else {
        // Generic tail (never taken for N = 851,968, a multiple of 64)
        for (int k = 0; k < 8; ++k) {
            int row = row0 + k;
            if (row < n) {
                int s = slot[ids[row]];
                const float4* src =
                    (const float4*)(emb + (size_t)s * EMB_DIM) + lane;
                float4* dst = (float4*)(out + (size_t)row * EMB_DIM) + lane;
                *dst = *src;
            }
        }
    }
}

extern "C" void kernel_launch(void* const* d_in, const int* in_sizes, int n_in,
                              void* d_out, int out_size, void* d_ws, size_t ws_size,
                              hipStream_t stream) {
    const int* ids   = (const int*)d_in[0];       // (4096,26,8) int32
    const float* emb = (const float*)d_in[1];     // (1e6,128) f32
    // d_in[2] = default_embedding: unreachable (n_unique <= 851,968 < 1e6)
    float* out = (float*)d_out;
    const int n = in_sizes[0];                    // 851,968 ids

    // Workspace layout (~4.07 MB)
    int* first_pos  = (int*)d_ws;                 // [500000]
    int* slot       = first_pos + RAW_ID_SPACE;   // [500000]
    int* block_sums = slot + RAW_ID_SPACE;        // [<=8192]
    int* block_off  = block_sums + 8192;          // [<=8192]

    const int nb = (n + 1023) / 1024;             // 832 scan blocks

    he_init_firstpos<<<(RAW_ID_SPACE + 255) / 256, 256, 0, stream>>>(
        first_pos, RAW_ID_SPACE);
    he_scatter_min<<<((n + 3) / 4 + 255) / 256, 256, 0, stream>>>(
        ids, first_pos, n);
    he_count_firsts<<<nb, 256, 0, stream>>>(ids, first_pos, block_sums, n);
    he_scan_blocksums<<<1, 1024, 0, stream>>>(block_sums, block_off, nb);
    he_write_slots<<<nb, 256, 0, stream>>>(ids, first_pos, block_off, slot, n);
    he_gather_rows<<<(n + 63) / 64, 256, 0, stream>>>(ids, slot, emb, out, n);
    (void)out_size; (void)ws_size; (void)n_in;
}